// AttentionPoolingLayer_68624987455577
// MI455X (gfx1250) — compile-verified
//
#include <hip/hip_runtime.h>
#include <math.h>

typedef __attribute__((ext_vector_type(2))) float v2f;
typedef __attribute__((ext_vector_type(8))) float v8f;

#define L_DIM     200
#define D_DIM     64
#define LPAD      208      // 13 * 16 WMMA M-tiles
#define KSTRIDE   68       // padded LDS row stride (floats) -> bank-conflict free
#define NTHREADS  256
#define NWAVES    8
#define NTILES    13

__global__ __launch_bounds__(NTHREADS)
void attn_pool_kernel(const float* __restrict__ query,
                      const float* __restrict__ keys,
                      const float* __restrict__ W,
                      float* __restrict__ out)
{
    __shared__ __align__(16) float sK[LPAD * KSTRIDE];  // keys tile, padded
    __shared__ float sQW[D_DIM];                        // query[b,:] * W[:]
    __shared__ float sS[LPAD];                          // scores -> weights
    __shared__ float sRed[NTHREADS];                    // reductions

    const int b    = blockIdx.x;
    const int t    = threadIdx.x;
    const int lane = t & 31;
    const int wave = t >> 5;

    // ---- Stage A: async-copy keys[b] (200x64 f32) directly into LDS ----
    // global_load_async_to_lds_b128: memory -> LDS, no VGPR round trip,
    // tracked by ASYNCcnt. VDST VGPR = workgroup-relative LDS byte offset
    // (= low 32 bits of the generic pointer to a __shared__ object).
    {
        const unsigned sKbase = (unsigned)(uintptr_t)&sK[0];
        const char* gbase = (const char*)(keys + (size_t)b * (L_DIM * D_DIM));
        for (int i = t; i < (L_DIM * D_DIM / 4); i += NTHREADS) {
            int l = i >> 4;              // 16 float4 per row
            int c = i & 15;
            unsigned lds_addr = sKbase + (unsigned)((l * KSTRIDE + 4 * c) * 4);
            const char* gptr = gbase + (size_t)i * 16;
            asm volatile("global_load_async_to_lds_b128 %0, %1, off"
                         :: "v"(lds_addr), "v"(gptr)
                         : "memory");
        }
        // zero the pad rows 200..207 (contiguous region) via normal DS stores
        for (int i = t; i < (LPAD - L_DIM) * KSTRIDE; i += NTHREADS)
            sK[L_DIM * KSTRIDE + i] = 0.0f;
        if (t < D_DIM) sQW[t] = query[(size_t)b * D_DIM + t] * W[t];
        asm volatile("s_wait_asynccnt 0x0" ::: "memory");
    }
    __syncthreads();

    // ---- Stage B: scores via V_WMMA_F32_16X16X4_F32 ----
    // S(200x1) = K_b(200x64) x qw(64x1); qw replicated over 16 B-columns.
    {
        const int off  = (lane < 16) ? 0 : 2;   // K-pair owned by this half-wave
        const int mrow = lane & 15;             // A-matrix row (M) for this lane
        for (int m = wave; m < NTILES; m += NWAVES) {
            const float* rowbase = &sK[(16 * m + mrow) * KSTRIDE];
            v8f acc = {0.f, 0.f, 0.f, 0.f, 0.f, 0.f, 0.f, 0.f};
            #pragma unroll
            for (int k = 0; k < 16; ++k) {
                const int d0 = 4 * k + off;
                v2f a  = *(const v2f*)(rowbase + d0);   // A: 16x4 tile of keys
                v2f bb = *(const v2f*)(&sQW[d0]);       // B: qw in every column
                acc = __builtin_amdgcn_wmma_f32_16x16x4_f32(
                        false, a, false, bb, (short)0, acc, false, false);
            }
            // C/D layout: lanes 0..15 hold M=0..7 in v[0..7], lanes 16..31 hold M=8..15.
            // All N columns identical -> lanes 0 and 16 publish the 16 scores.
            if (lane == 0 || lane == 16) {
                int l0 = 16 * m + ((lane == 16) ? 8 : 0);
                #pragma unroll
                for (int r = 0; r < 8; ++r) {
                    int l = l0 + r;
                    if (l < L_DIM) sS[l] = tanhf(acc[r]);
                }
            }
        }
    }
    __syncthreads();

    // ---- Stage C: softmax over L (in LDS) ----
    float sc = (t < L_DIM) ? sS[t] : -INFINITY;
    sRed[t] = sc;
    __syncthreads();
    for (int s = NTHREADS / 2; s > 0; s >>= 1) {
        if (t < s) sRed[t] = fmaxf(sRed[t], sRed[t + s]);
        __syncthreads();
    }
    float mx = sRed[0];
    __syncthreads();
    float e = (t < L_DIM) ? __expf(sc - mx) : 0.0f;
    sRed[t] = e;
    __syncthreads();
    for (int s = NTHREADS / 2; s > 0; s >>= 1) {
        if (t < s) sRed[t] += sRed[t + s];
        __syncthreads();
    }
    float inv = 1.0f / sRed[0];
    __syncthreads();
    if (t < L_DIM) sS[t] = e * inv;   // normalized weights
    __syncthreads();

    // ---- Stage D: out[b,d] = sum_l w[l] * K[l,d] (keys reused from LDS) ----
    const int d = t & (D_DIM - 1);
    const int g = t >> 6;             // 4 groups x 50 l's
    float acc = 0.0f;
    for (int l = g * 50; l < g * 50 + 50; ++l)
        acc += sS[l] * sK[l * KSTRIDE + d];
    sRed[t] = acc;
    __syncthreads();
    if (t < D_DIM)
        out[(size_t)b * D_DIM + d] =
            sRed[d] + sRed[d + 64] + sRed[d + 128] + sRed[d + 192];
}

extern "C" void kernel_launch(void* const* d_in, const int* in_sizes, int n_in,
                              void* d_out, int out_size, void* d_ws, size_t ws_size,
                              hipStream_t stream) {
    const float* query = (const float*)d_in[0];   // [B, 64]
    const float* keys  = (const float*)d_in[1];   // [B, 200, 64]
    const float* W     = (const float*)d_in[2];   // [1, 64]
    float* out         = (float*)d_out;           // [B, 64]

    const int B = in_sizes[0] / D_DIM;            // 16384
    dim3 grid(B), block(NTHREADS);
    hipLaunchKernelGGL(attn_pool_kernel, grid, block, 0, stream,
                       query, keys, W, out);
}